// BlockLinear_90752658965115
// MI455X (gfx1250) — compile-verified
//
#include <hip/hip_runtime.h>

// Block-diagonal GEMM: out[k*256+m, b] = sum_j blocks[k][m][j] * inp[k*256+j, b] + bias[k*256+m]
// NUM_BLOCKS=8, D=256, B=8192, all f32.  WMMA f32 16x16x4, async global->LDS double buffering.
// Per workgroup: full 256-row block x 128 columns. Per wave: 2x8 tiles of 16x16.

#define NUM_BLOCKS 8
#define D 256
#define BCOLS 8192
#define CT 128              // output columns per workgroup
#define KC 16               // K chunk staged in LDS (16 chunks total)
#define WSTRIDE (KC + 4)    // 20 floats: row base stays 16B aligned; 20*m mod 64 injective for m=0..15
#define NCHUNKS (D / KC)

typedef float v2f __attribute__((ext_vector_type(2)));
typedef float v4f __attribute__((ext_vector_type(4)));
typedef float v8f __attribute__((ext_vector_type(8)));

#if defined(__gfx1250__) && __has_builtin(__builtin_amdgcn_global_load_async_to_lds_b128)
#define HAVE_ASYNC_LDS 1
#else
#define HAVE_ASYNC_LDS 0
#endif

#if HAVE_ASYNC_LDS
#if __has_builtin(__builtin_amdgcn_s_wait_asynccnt)
#define WAIT_ASYNC() __builtin_amdgcn_s_wait_asynccnt(0)
#else
#define WAIT_ASYNC() asm volatile("s_wait_asynccnt 0x0" ::: "memory")
#endif
// Builtin signature (probe-confirmed via diagnostic): pointers to
// 'int __attribute__((vector_size(16)))' in global (AS1) / shared (AS3) address spaces.
typedef int v4i __attribute__((vector_size(16)));
typedef __attribute__((address_space(1))) v4i as1_v4i;
typedef __attribute__((address_space(3))) v4i as3_v4i;
#define ASYNC_CP_B128(gptr, lptr) \
    __builtin_amdgcn_global_load_async_to_lds_b128((as1_v4i*)(gptr), (as3_v4i*)(lptr), 0, 0)
#else
#define WAIT_ASYNC() ((void)0)
#endif

__global__ __launch_bounds__(256) void block_linear_wmma_f32(
    const float* __restrict__ inp,     // (2048, 8192)
    const float* __restrict__ blocks,  // (8, 256, 256)
    const float* __restrict__ bias,    // (2048,)
    float* __restrict__ out)           // (2048, 8192)
{
    __shared__ float ldsW[2 * D * WSTRIDE];  // double-buffered W chunk: 2 x 256 x 20  (40 KB)
    __shared__ float ldsX[2 * KC * CT];      // double-buffered X chunk: 2 x 16 x 128  (16 KB)

    const int k    = blockIdx.y;
    const int col0 = blockIdx.x * CT;
    const int tid  = threadIdx.x;
    const int lane = tid & 31;
    const int wave = tid >> 5;        // 8 waves
    const int half = lane >> 4;
    const int l16  = lane & 15;

    const float* Wk = blocks + (size_t)k * D * D;
    const float* Xk = inp    + (size_t)k * D * BCOLS;

    const int m0 = wave * 32;         // wave's 32-row slice of the 256-row block

    v8f acc[2][8];
    #pragma unroll
    for (int i = 0; i < 2; ++i)
        #pragma unroll
        for (int j = 0; j < 8; ++j)
            #pragma unroll
            for (int e = 0; e < 8; ++e)
                acc[i][j][e] = 0.0f;

    // ---- chunk staging: global -> LDS (async B128 if available, else sync) ----
    // W chunk: 256 rows x 16 floats (64B/row = 4 x b128). 1024 segments -> 4 wave-instr/wave.
    // X chunk: 16 rows x 128 floats (512B/row = 32 x b128). 512 segments -> 2 wave-instr/wave.
    auto stage = [&](int buf, int kc) {
        float* wb = ldsW + buf * (D * WSTRIDE);
        float* xb = ldsX + buf * (KC * CT);
#if HAVE_ASYNC_LDS
        #pragma unroll
        for (int i = 0; i < 4; ++i) {
            const int s   = (wave * 4 + i) * 32 + lane;  // 0..1023
            const int m   = s >> 2;
            const int seg = (s & 3) * 4;
            const float* g = Wk + (size_t)m * D + kc + seg;
            float*       l = wb + m * WSTRIDE + seg;
            ASYNC_CP_B128(g, l);
        }
        #pragma unroll
        for (int i = 0; i < 2; ++i) {
            const int s   = (wave * 2 + i) * 32 + lane;  // 0..511
            const int r   = s >> 5;
            const int seg = (s & 31) * 4;
            const float* g = Xk + (size_t)(kc + r) * BCOLS + col0 + seg;
            float*       l = xb + r * CT + seg;
            ASYNC_CP_B128(g, l);
        }
#else
        #pragma unroll
        for (int i = 0; i < 4; ++i) {
            const int s   = (wave * 4 + i) * 32 + lane;
            const int m   = s >> 2;
            const int seg = (s & 3) * 4;
            v4f w = *(const v4f*)(Wk + (size_t)m * D + kc + seg);
            *(v4f*)(wb + m * WSTRIDE + seg) = w;   // 16B-aligned (row base = m*80B)
        }
        #pragma unroll
        for (int i = 0; i < 2; ++i) {
            const int s   = (wave * 2 + i) * 32 + lane;
            const int r   = s >> 5;
            const int seg = (s & 31) * 4;
            v4f x = *(const v4f*)(Xk + (size_t)(kc + r) * BCOLS + col0 + seg);
            *(v4f*)(xb + r * CT + seg) = x;
        }
#endif
    };

    // Prologue: stage chunk 0, wait, barrier.
    stage(0, 0);
    WAIT_ASYNC();
    __syncthreads();

    for (int c = 0; c < NCHUNKS; ++c) {
        const int buf = c & 1;
        if (c + 1 < NCHUNKS)
            stage(buf ^ 1, (c + 1) * KC);   // prefetch next chunk while computing

        const float* wb = ldsW + buf * (D * WSTRIDE);
        const float* xb = ldsX + buf * (KC * CT);

        // ---- Compute: 4 K-steps of 4 via V_WMMA_F32_16X16X4_F32 ----
        #pragma unroll
        for (int k0 = 0; k0 < KC; k0 += 4) {
            const int kk = k0 + 2 * half;   // lanes 16-31 take K+2 (ISA 7.12.2)

            // A fragments (16x4): lane = M row; (kk, kk+1) adjacent & 8B aligned -> b64 load
            v2f a[2];
            #pragma unroll
            for (int i = 0; i < 2; ++i) {
                const int m = m0 + i * 16 + l16;
                a[i] = *(const v2f*)(wb + m * WSTRIDE + kk);
            }
            // B fragments (4x16): lane = N col
            v2f b[8];
            #pragma unroll
            for (int j = 0; j < 8; ++j) {
                const int n = j * 16 + l16;
                b[j][0] = xb[kk * CT + n];
                b[j][1] = xb[(kk + 1) * CT + n];
            }

            #pragma unroll
            for (int i = 0; i < 2; ++i)
                #pragma unroll
                for (int j = 0; j < 8; ++j)
                    acc[i][j] = __builtin_amdgcn_wmma_f32_16x16x4_f32(
                        false, a[i], false, b[j],
                        (short)0, acc[i][j], false, false);
        }

        WAIT_ASYNC();        // next chunk's async copies have landed (own wave's)
        __syncthreads();     // ...and everyone else's; also fences LDS reuse
    }

    // ---- Epilogue: bias add + store ----
    // C/D layout: VGPR v holds M = v (lanes 0-15) / v+8 (lanes 16-31), N = lane&15.
    const float* biask = bias + k * D;
    #pragma unroll
    for (int i = 0; i < 2; ++i) {
        #pragma unroll
        for (int j = 0; j < 8; ++j) {
            const int n = col0 + j * 16 + l16;
            #pragma unroll
            for (int v = 0; v < 8; ++v) {
                const int m = m0 + i * 16 + v + 8 * half;
                out[(size_t)(k * D + m) * BCOLS + n] = acc[i][j][v] + biask[m];
            }
        }
    }
}

extern "C" void kernel_launch(void* const* d_in, const int* in_sizes, int n_in,
                              void* d_out, int out_size, void* d_ws, size_t ws_size,
                              hipStream_t stream) {
    (void)in_sizes; (void)n_in; (void)out_size; (void)d_ws; (void)ws_size;
    const float* inp    = (const float*)d_in[0];
    const float* blocks = (const float*)d_in[1];
    const float* bias   = (const float*)d_in[2];
    float* out = (float*)d_out;

    dim3 grid(BCOLS / CT, NUM_BLOCKS);  // (64, 8) workgroups
    dim3 block(256);                    // 8 waves (wave32)
    block_linear_wmma_f32<<<grid, block, 0, stream>>>(inp, blocks, bias, out);
}